// EquivariantGraphConv_15324443312657
// MI455X (gfx1250) — compile-verified
//
#include <hip/hip_runtime.h>
#include <stdint.h>

typedef float v2f __attribute__((ext_vector_type(2)));
typedef float v4f __attribute__((ext_vector_type(4)));
typedef float v8f __attribute__((ext_vector_type(8)));

#define BT        16            // batches per workgroup -> 272 rows = 17 M-tiles
#define NNODE     17
#define FIN       64
#define FOUT      64
#define KTOT      128           // concat [x | adj@x]
#define ROWSTRIDE 132           // LDS row stride (floats), 132%64=4 -> bank-conflict-free
#define ROWS      (BT * NNODE)  // 272
#define XU_FLOATS (ROWS * ROWSTRIDE)       // 35904
#define WT_FLOATS (FOUT * ROWSTRIDE)       // 8448
#define ADJ_STRIDE 20
#define ADJ_FLOATS (32 * ADJ_STRIDE)       // 640 (zero-padded 32x20)
#define SMEM_FLOATS (XU_FLOATS + WT_FLOATS + ADJ_FLOATS + FOUT)

__global__ __launch_bounds__(256)
void egc_wmma_kernel(const float* __restrict__ x,
                     const float* __restrict__ adj,
                     const float* __restrict__ wself,
                     const float* __restrict__ wneigh,
                     const float* __restrict__ bias,
                     float* __restrict__ out)
{
  extern __shared__ __align__(16) float smem[];
  float* xu    = smem;                    // [272][132]: cols 0..63 = x, 64..127 = adj@x
  float* WT    = smem + XU_FLOATS;        // [64][132]:  WT[o][k] = stacked W[k][o]
  float* adjp  = WT + WT_FLOATS;          // [32][20] zero-padded adjacency
  float* biasS = adjp + ADJ_FLOATS;       // [64]

  const int tid  = threadIdx.x;
  const int lane = tid & 31;              // wave32
  const int wave = tid >> 5;
  const int l15  = lane & 15;
  const int kh   = lane >> 4;             // K-half selector in 16x16x4 fragments
  const long wg  = blockIdx.x;

  // ---------------- Phase 1: stage operands into LDS ----------------
  {
    const float* xg = x + wg * (long)(BT * NNODE * FIN);
    // x: 16 batches * 17 * 64 floats = 4352 float4 chunks, contiguous in global.
    // Use CDNA5 async direct-to-LDS copy (no VGPR round trip, tracked by ASYNCcnt).
    for (int g = tid; g < (BT * NNODE * FIN) / 4; g += 256) {
      const int row = g >> 4;             // 16 float4 per 64-float row
      const int col = (g & 15) << 2;
      const uint32_t ldsOff =
          (uint32_t)(uintptr_t)(xu + row * ROWSTRIDE + col);   // low 32b = LDS offset
      const unsigned long long gaddr = (unsigned long long)(uintptr_t)(xg + g * 4);
      asm volatile("global_load_async_to_lds_b128 %0, %1, off"
                   :: "v"(ldsOff), "v"(gaddr) : "memory");
    }
    // W^T: WT[o][k] with k<64 -> Wself[k][o], k>=64 -> Wneigh[k-64][o]
    for (int idx = tid; idx < KTOT * FOUT; idx += 256) {
      const int k = idx >> 6;
      const int o = idx & 63;
      const float w = (k < FIN) ? wself[k * FOUT + o] : wneigh[(k - FIN) * FOUT + o];
      WT[o * ROWSTRIDE + k] = w;
    }
    // zero-padded adjacency 32x20
    for (int idx = tid; idx < ADJ_FLOATS; idx += 256) {
      const int n = idx / ADJ_STRIDE;
      const int m = idx % ADJ_STRIDE;
      adjp[idx] = (n < NNODE && m < NNODE) ? adj[n * NNODE + m] : 0.0f;
    }
    if (tid < FOUT) biasS[tid] = bias[tid];
    // async LDS fill must land before the barrier releases consumers
    asm volatile("s_wait_asynccnt 0x0" ::: "memory");
  }
  __syncthreads();

  // ---- Phase 2: xn[b] = adj @ x[b] via v_wmma_f32_16x16x4_f32 -> xu[:,64:128]
  // 128 tiles: b(16) x mtile(2) x ntile(4) -> 16 tiles/wave, uniform per wave
  for (int t = wave; t < BT * 2 * 4; t += 8) {
    const int b  = t >> 3;
    const int mt = (t >> 2) & 1;
    const int nt = t & 3;
    float* xb = xu + b * (NNODE * ROWSTRIDE);
    const float* arow = adjp + (mt * 16 + l15) * ADJ_STRIDE;  // A row = output node n
    const int col = nt * 16 + l15;
    v8f acc = {};
#pragma unroll
    for (int kb = 0; kb < 20; kb += 4) {
      const int k0 = kb + 2 * kh;
      v2f a = *(const v2f*)(arow + k0);                 // contiguous (k,k+1) pair
      // B frag = x[b][k][col]; clamp+mask (no branch -> EXEC stays all-ones)
      const int k0c = (k0     < NNODE) ? k0       : (NNODE - 1);
      const int k1c = (k0 + 1 < NNODE) ? (k0 + 1) : (NNODE - 1);
      v2f bf;
      bf.x = xb[k0c * ROWSTRIDE + col] * ((k0     < NNODE) ? 1.0f : 0.0f);
      bf.y = xb[k1c * ROWSTRIDE + col] * ((k0 + 1 < NNODE) ? 1.0f : 0.0f);
      acc = __builtin_amdgcn_wmma_f32_16x16x4_f32(false, a, false, bf,
                                                  (short)0, acc, false, false);
    }
    if (mt == 0) {
      // all 16 rows valid: unconditional stores (C layout: M = r + 8*kh)
#pragma unroll
      for (int r = 0; r < 8; ++r)
        xb[(r + 8 * kh) * ROWSTRIDE + FIN + col] = acc[r];
    } else {
      // only row n=16 valid: it is acc[0] of lanes 0..15
      if (kh == 0) xb[16 * ROWSTRIDE + FIN + col] = acc[0];
    }
  }
  __syncthreads();

  // ---- Phase 3: out = [x|xn] @ [Ws;Wn] + bias, K = 128
  // 68 tiles: mtile(17) x ntile(4)
  for (int t = wave; t < 17 * 4; t += 8) {
    const int mt = t >> 2;
    const int nt = t & 3;
    const float* arow = xu + (mt * 16 + l15) * ROWSTRIDE + 2 * kh;
    const float* brow = WT + (nt * 16 + l15) * ROWSTRIDE + 2 * kh;
    v8f acc = {};
#pragma unroll
    for (int kb = 0; kb < KTOT; kb += 4) {
      v2f a  = *(const v2f*)(arow + kb);   // single ds_load_b64 each
      v2f bf = *(const v2f*)(brow + kb);
      acc = __builtin_amdgcn_wmma_f32_16x16x4_f32(false, a, false, bf,
                                                  (short)0, acc, false, false);
    }
    const int col = nt * 16 + l15;
    const float bv = biasS[col];
    const long rowbase = wg * (long)ROWS + mt * 16 + 8 * kh;
#pragma unroll
    for (int r = 0; r < 8; ++r) {
      out[(rowbase + r) * (long)FOUT + col] = acc[r] + bv;
    }
  }
}

extern "C" void kernel_launch(void* const* d_in, const int* in_sizes, int n_in,
                              void* d_out, int out_size, void* d_ws, size_t ws_size,
                              hipStream_t stream) {
  const float* x     = (const float*)d_in[0];
  const float* adj   = (const float*)d_in[1];
  const float* ws    = (const float*)d_in[2];
  const float* wn    = (const float*)d_in[3];
  const float* bias  = (const float*)d_in[4];
  float* out = (float*)d_out;

  const int B = 65536;
  dim3 grid(B / BT), block(256);
  egc_wmma_kernel<<<grid, block, SMEM_FLOATS * sizeof(float), stream>>>(
      x, adj, ws, wn, bias, out);
}